// DynamicDilationUnfold_57595511439437
// MI455X (gfx1250) — compile-verified
//
#include <hip/hip_runtime.h>
#include <hip/hip_bf16.h>

namespace {
constexpr int Bn = 4, Cn = 64, Hn = 128, Wn = 128, Kn = 3, Gn = 4, Cg = 16;
constexpr int PADc = 1;
constexpr int ROWS  = 4;          // output rows per block
constexpr int LROWS = ROWS + 6;   // staged input rows: halo [-1 .. +5] around band
constexpr int HoW   = Hn * Wn;    // 16384
constexpr int NT    = 256;        // threads per block (8 waves of 32)
}

__global__ __launch_bounds__(NT)
void ddunfold_kernel(const float* __restrict__ x,
                     const float* __restrict__ dmap,
                     float* __restrict__ out)
{
    // Layout [lrow][c][col]: channel stride = 128 floats = 512B (fits DS imm offset
    // for all 16 channels). 10*16*128*4B = 80 KB.
    __shared__ float lds[LROWS * Cg * Wn];

    const int tid  = threadIdx.x;
    const int lane = tid & 31;
    const int warp = tid >> 5;
    const int band = blockIdx.x & 31;        // Ho/ROWS = 32
    const int g    = (blockIdx.x >> 5) & 3;
    const int b    = blockIdx.x >> 7;
    const int r0   = band * ROWS;

    // ---------------- stage input band into LDS via async global->LDS DMA ----
    // 160 (lrow,c) pairs * 32 lanes * 16B = 80KB. p = warp + j*8 covers 0..159.
    const float* src = x + (size_t)(b * Cn + g * Cg) * HoW;
    #pragma unroll
    for (int j = 0; j < (LROWS * Cg) / 8; ++j) {    // 20 async b128 per thread
        int p    = warp + j * 8;                    // 0..159
        int c    = p & 15;
        int lrow = p >> 4;                          // 0..9
        int gr   = r0 - 1 + lrow;
        gr = gr < 0 ? 0 : (gr > Hn - 1 ? Hn - 1 : gr);   // clamp (dup rows at edges)
        const float* gp = src + (size_t)c * HoW + gr * Wn + lane * 4;
        unsigned ldsOff = (unsigned)((((lrow * Cg + c) * Wn) + lane * 4) * 4);
        // Third operand: escape the LDS array so the "memory" clobber is known
        // to alias it (otherwise the LDS reads below fold to undef).
        asm volatile("global_load_async_to_lds_b128 %0, %1, off"
                     :: "v"(ldsOff), "v"(gp), "v"((const void*)lds) : "memory");
    }
    asm volatile("s_wait_asynccnt 0" ::: "memory");
    __syncthreads();

    // ---------------- gather / bilinear / scatter ----------------------------
    float* outg = out + (size_t)(b * Cn + g * Cg) * 9 * HoW;

    for (int it = 0; it < (ROWS * Wn) / NT; ++it) {   // 2 pixels per thread
        int pos = tid + it * NT;
        int ho  = r0 + (pos >> 7);
        int wo  = pos & 127;

        float d = dmap[(((b * Gn + g) * Hn) + ho) * Wn + wo];
        float* op = outg + ho * Wn + wo;

        #pragma unroll
        for (int kh = 0; kh < Kn; ++kh) {
            // separate-rounded mul+add to match reference floor() decisions
            float ph  = __fadd_rn((float)(ho - PADc), __fmul_rn((float)kh, d));
            float h0f = floorf(ph);
            float lh  = ph - h0f;
            int   ih0 = (int)h0f;                     // >= -1 always
            bool  hv0 = (ih0 >= 0) && (ih0 <= Hn - 1);
            bool  hv1 = (ih0 + 1 <= Hn - 1);          // ih0+1 >= 0 always
            int   rr0 = ih0 < 0 ? 0 : (ih0 > Hn - 1 ? Hn - 1 : ih0);
            int   rr1 = ih0 + 1 > Hn - 1 ? Hn - 1 : ih0 + 1;
            int   lb0 = (rr0 - (r0 - 1)) * (Cg * Wn); // lds row base, in [0,10)*2048
            int   lb1 = (rr1 - (r0 - 1)) * (Cg * Wn);

            #pragma unroll
            for (int kw = 0; kw < Kn; ++kw) {
                float pw  = __fadd_rn((float)(wo - PADc), __fmul_rn((float)kw, d));
                float w0f = floorf(pw);
                float lw  = pw - w0f;
                int   iw0 = (int)w0f;
                bool  wv0 = (iw0 >= 0) && (iw0 <= Wn - 1);
                bool  wv1 = (iw0 + 1 <= Wn - 1);
                int   cc0 = iw0 < 0 ? 0 : (iw0 > Wn - 1 ? Wn - 1 : iw0);
                int   cc1 = iw0 + 1 > Wn - 1 ? Wn - 1 : iw0 + 1;

                float w00 = (1.f - lh) * (1.f - lw); if (!(hv0 && wv0)) w00 = 0.f;
                float w01 = (1.f - lh) * lw;         if (!(hv0 && wv1)) w01 = 0.f;
                float w10 = lh * (1.f - lw);         if (!(hv1 && wv0)) w10 = 0.f;
                float w11 = lh * lw;                 if (!(hv1 && wv1)) w11 = 0.f;

                int o00 = lb0 + cc0, o01 = lb0 + cc1;   // + c*Wn selects channel
                int o10 = lb1 + cc0, o11 = lb1 + cc1;
                const int kidx = kh * Kn + kw;

                #pragma unroll
                for (int c = 0; c < Cg; ++c) {
                    const int cb = c * Wn;              // 512B steps: DS imm offset
                    float v = w00 * lds[o00 + cb] + w01 * lds[o01 + cb]
                            + w10 * lds[o10 + cb] + w11 * lds[o11 + cb];
                    __builtin_nontemporal_store(v, op + (size_t)(c * 9 + kidx) * HoW);
                }
            }
        }
    }
}

extern "C" void kernel_launch(void* const* d_in, const int* in_sizes, int n_in,
                              void* d_out, int out_size, void* d_ws, size_t ws_size,
                              hipStream_t stream) {
    const float* x    = (const float*)d_in[0];   // (4,64,128,128) f32
    const float* dmap = (const float*)d_in[1];   // (4,4,128,128)  f32
    float* out = (float*)d_out;                  // (4,576,16384)  f32
    (void)in_sizes; (void)n_in; (void)out_size; (void)d_ws; (void)ws_size;
    dim3 grid(Bn * Gn * (Hn / ROWS));            // 512 blocks
    ddunfold_kernel<<<grid, NT, 0, stream>>>(x, dmap, out);
}